// _GraphSAGE_18279380812446
// MI455X (gfx1250) — compile-verified
//
#include <hip/hip_runtime.h>

typedef __attribute__((ext_vector_type(2))) float v2f;
typedef __attribute__((ext_vector_type(8))) float v8f;

#define DIM   256
#define N1C   40000
#define N2C   8000
#define E1C   1000000
#define E2C   200000
#define LROW  (DIM + 4)   // padded LDS row stride (floats) -> bank-conflict free

// ---------------------------------------------------------------------------
// Zero int counters (off1 | off2 | cur1 | cur2 contiguous).
// ---------------------------------------------------------------------------
__global__ void zero_int(int* __restrict__ p, int n) {
    int i = blockIdx.x * blockDim.x + threadIdx.x;
    if (i < n) p[i] = 0;
}

// ---------------------------------------------------------------------------
// Per-destination in-degree histogram (int atomics: E ops total, cheap).
// ---------------------------------------------------------------------------
__global__ void edge_hist(const int* __restrict__ dst, int* __restrict__ cnt,
                          int n) {
    int i = blockIdx.x * blockDim.x + threadIdx.x;
    if (i < n) atomicAdd(cnt + dst[i], 1);
}

// ---------------------------------------------------------------------------
// Single-block exclusive scan over n counts (in-place); data[n] = total.
// n <= 40000 -> 40 chunks of 1024, Hillis-Steele in LDS. Negligible runtime.
// ---------------------------------------------------------------------------
__global__ __launch_bounds__(1024)
void scan_excl(int* __restrict__ data, int n) {
    __shared__ int tmp[1024];
    __shared__ int running;
    int tid = threadIdx.x;
    if (tid == 0) running = 0;
    __syncthreads();
    for (int base = 0; base < n; base += 1024) {
        int i = base + tid;
        int v = (i < n) ? data[i] : 0;
        tmp[tid] = v;
        __syncthreads();
        for (int d = 1; d < 1024; d <<= 1) {
            int t = (tid >= d) ? tmp[tid - d] : 0;
            __syncthreads();
            tmp[tid] += t;
            __syncthreads();
        }
        int incl = tmp[tid];
        int r = running;
        if (i < n) data[i] = r + incl - v;   // exclusive
        __syncthreads();
        if (tid == 1023) running = r + incl; // chunk total
        __syncthreads();
    }
    if (tid == 0) data[n] = running;
}

// ---------------------------------------------------------------------------
// Fill CSR adjacency: eid[off[d] + k] = src of k-th edge landing on d.
// ---------------------------------------------------------------------------
__global__ void edge_fill(const int* __restrict__ src,
                          const int* __restrict__ dst,
                          const int* __restrict__ off,
                          int* __restrict__ cur,
                          int* __restrict__ eid, int n) {
    int i = blockIdx.x * blockDim.x + threadIdx.x;
    if (i < n) {
        int d = dst[i];
        int p = atomicAdd(cur + d, 1);
        eid[off[d] + p] = src[i];
    }
}

// ---------------------------------------------------------------------------
// Fused SAGE layer: out[i,:] = tanh(mean[i,:] @ Wl^T + b + x[i,:] @ Wr^T)
//
// Block = 512 threads = 16 waves, one block per 16-row output stripe.
//  Phase 1: wave w gathers the neighbor-mean of row m0+w straight into LDS
//           (neighbor src ids broadcast across the wave with __shfl; each
//           lane accumulates 8 floats via two float4 loads per neighbor).
//           The block co-stages the 16x256 x_tgt tile.
//  Phase 2: each wave owns one 16-col tile; K=256 loop with
//           V_WMMA_F32_16X16X4_F32 (2 WMMAs per k-step: Wl & Wr parts),
//           A from LDS (padded stride), B from the L2-hot 256KB weights.
//
// Fragment layouts per CDNA5 ISA 7.12.2:
//   A (16x4 f32): lane l holds row M=l%16, K = 2*(l/16)+{0,1}
//   B (4x16 f32): lane l holds col N=l%16, K = 2*(l/16)+{0,1}
//   C/D (16x16):  VGPR r holds M = r + 8*(l/16), N = l%16
// ---------------------------------------------------------------------------
__global__ __launch_bounds__(512)
void sage_layer(const float* __restrict__ x,    // source features (n_src x 256)
                const int* __restrict__ off,    // CSR offsets (M+1)
                const int* __restrict__ eid,    // src ids grouped by dst
                const float* __restrict__ Wl,   // (256,256) row-major
                const float* __restrict__ bias, // (256)
                const float* __restrict__ Wr,   // (256,256) row-major
                float* __restrict__ out,        // (M,256)
                int M) {
    __shared__ float lmean[16 * LROW];
    __shared__ float lx[16 * LROW];

    const int m0   = blockIdx.x * 16;
    const int tid  = threadIdx.x;
    const int lane = tid & 31;
    const int wv   = tid >> 5;     // 0..15

    // --- stage x_tgt tile (coalesced b32) ---
    for (int i = tid; i < 16 * DIM; i += 512) {
        int row = i >> 8;
        int col = i & (DIM - 1);
        lx[row * LROW + col] = x[(long)(m0 + row) * DIM + col];
    }

    // --- wave wv gathers + means row m0+wv into LDS ---
    {
        int row = m0 + wv;
        int beg = off[row];
        int end = off[row + 1];
        float4 a0 = make_float4(0.f, 0.f, 0.f, 0.f);
        float4 a1 = make_float4(0.f, 0.f, 0.f, 0.f);
        for (int e = beg; e < end; e += 32) {
            int myid = (e + lane < end) ? eid[e + lane] : 0;
            int cnt = min(32, end - e);
            for (int j = 0; j < cnt; ++j) {
                int s = __shfl(myid, j, 32);
                const float4* xr = (const float4*)(x + (long)s * DIM);
                float4 u = xr[lane];        // floats [lane*4 .. +3]
                float4 v = xr[lane + 32];   // floats [128+lane*4 .. +3]
                a0.x += u.x; a0.y += u.y; a0.z += u.z; a0.w += u.w;
                a1.x += v.x; a1.y += v.y; a1.z += v.z; a1.w += v.w;
            }
        }
        float inv = 1.0f / fmaxf((float)(end - beg), 1.0f);
        float* mrow = lmean + wv * LROW;
        mrow[lane * 4 + 0] = a0.x * inv;
        mrow[lane * 4 + 1] = a0.y * inv;
        mrow[lane * 4 + 2] = a0.z * inv;
        mrow[lane * 4 + 3] = a0.w * inv;
        mrow[128 + lane * 4 + 0] = a1.x * inv;
        mrow[128 + lane * 4 + 1] = a1.y * inv;
        mrow[128 + lane * 4 + 2] = a1.z * inv;
        mrow[128 + lane * 4 + 3] = a1.w * inv;
    }
    __syncthreads();

    // --- WMMA K-loop ---
    const int half = lane >> 4;    // K sub-pair select
    const int lrow = lane & 15;    // M (A) / N (B,C,D)
    const int n0   = wv * 16;

    const float* wl_r = Wl + (long)(n0 + lrow) * DIM;  // B[k][n] = Wl[n][k]
    const float* wr_r = Wr + (long)(n0 + lrow) * DIM;
    const float* lm_r = lmean + lrow * LROW;
    const float* lx_r = lx + lrow * LROW;

    v8f acc = {};
#pragma unroll 4
    for (int k = 0; k < DIM; k += 4) {
        int kk = k + 2 * half;
        v2f am = *(const v2f*)(lm_r + kk);   // ds_load_b64
        v2f bl = *(const v2f*)(wl_r + kk);   // global_load_b64
        acc = __builtin_amdgcn_wmma_f32_16x16x4_f32(
            false, am, false, bl, (short)0, acc, false, false);
        v2f ax = *(const v2f*)(lx_r + kk);
        v2f br = *(const v2f*)(wr_r + kk);
        acc = __builtin_amdgcn_wmma_f32_16x16x4_f32(
            false, ax, false, br, (short)0, acc, false, false);
    }

    float bb = bias[n0 + lrow];
    float* orow = out + (long)(m0 + half * 8) * DIM + n0 + lrow;
#pragma unroll
    for (int r = 0; r < 8; ++r) {
        orow[(long)r * DIM] = tanhf(acc[r] + bb);
    }
}

// ---------------------------------------------------------------------------
// Workspace layout (ints then floats, ~46 MB total):
//   off1 : N1+1      off2 : N2+1      cur1 : N1      cur2 : N2
//   eid1 : E1        eid2 : E2        h1   : N1*256 f32 (16B aligned)
// ---------------------------------------------------------------------------
extern "C" void kernel_launch(void* const* d_in, const int* in_sizes, int n_in,
                              void* d_out, int out_size, void* d_ws, size_t ws_size,
                              hipStream_t stream) {
    const float* nodes = (const float*)d_in[0];
    const float* W_l1  = (const float*)d_in[1];
    const float* b1    = (const float*)d_in[2];
    const float* W_r1  = (const float*)d_in[3];
    const float* W_l2  = (const float*)d_in[4];
    const float* b2    = (const float*)d_in[5];
    const float* W_r2  = (const float*)d_in[6];
    const int*   src1  = (const int*)d_in[7];
    const int*   dst1  = (const int*)d_in[8];
    const int*   src2  = (const int*)d_in[9];
    const int*   dst2  = (const int*)d_in[10];

    int* wsi  = (int*)d_ws;
    int* off1 = wsi;
    int* off2 = off1 + (N1C + 1);
    int* cur1 = off2 + (N2C + 1);
    int* cur2 = cur1 + N1C;
    int* eid1 = cur2 + N2C;
    int* eid2 = eid1 + E1C;
    long ihdr = (long)(N1C + 1) + (N2C + 1) + N1C + N2C + E1C + E2C;
    ihdr = (ihdr + 3) & ~3L;                       // 16B-align the float region
    float* h1 = (float*)(wsi + ihdr);
    float* h2 = (float*)d_out;

    const int nzero = (N1C + 1) + (N2C + 1) + N1C + N2C;  // off+cur regions

    // 1. Zero counters/cursors.
    zero_int<<<(nzero + 255) / 256, 256, 0, stream>>>(off1, nzero);
    // 2. In-degree histograms (counts accumulate into the off arrays).
    edge_hist<<<(E1C + 255) / 256, 256, 0, stream>>>(dst1, off1, E1C);
    edge_hist<<<(E2C + 255) / 256, 256, 0, stream>>>(dst2, off2, E2C);
    // 3. Exclusive scans -> CSR offsets.
    scan_excl<<<1, 1024, 0, stream>>>(off1, N1C);
    scan_excl<<<1, 1024, 0, stream>>>(off2, N2C);
    // 4. Fill CSR adjacency (src ids grouped by destination).
    edge_fill<<<(E1C + 255) / 256, 256, 0, stream>>>(src1, dst1, off1, cur1,
                                                     eid1, E1C);
    edge_fill<<<(E2C + 255) / 256, 256, 0, stream>>>(src2, dst2, off2, cur2,
                                                     eid2, E2C);
    // 5. Layer 1: gather-mean + WMMA GEMMs + bias + tanh -> h1 (40000 x 256).
    sage_layer<<<N1C / 16, 512, 0, stream>>>(nodes, off1, eid1,
                                             W_l1, b1, W_r1, h1, N1C);
    // 6. Layer 2: same over h1 -> d_out (8000 x 256).
    sage_layer<<<N2C / 16, 512, 0, stream>>>(h1, off2, eid2,
                                             W_l2, b2, W_r2, h2, N2C);
}